// Kenn_36077725286759
// MI455X (gfx1250) — compile-verified
//
#include <hip/hip_runtime.h>

typedef __attribute__((ext_vector_type(2))) float v2f;
typedef __attribute__((ext_vector_type(4))) float v4f;
typedef __attribute__((ext_vector_type(8))) float v8f;

#define NPAD 64   // hidden 50 padded to 64 (4 WMMA col-tiles)
#define NC   6    // classes
#define ZS   8    // z row stride (padded for b128 gathers)

// ---------------------------------------------------------------------------
// fp32 WMMA GEMM with LDS-staged, double-buffered B.
//   out[M x out_cols] = relu?(A[M x K] @ B + bias)
// B quad-packed: Bq[(k/4)*Ncols*4 + n*4 + j] = B[k+j][n]  (zero-padded), so a
// K-chunk is a contiguous span; the block cooperatively copies it to LDS, and
// all 8 waves read B fragments with ds_load_b128 (8x cut in L2 B-traffic).
// K-slot mapping per 8-k step (A and B agree):
//   WMMA0: logical {k, k+1, k+4, k+5}; WMMA1: {k+2, k+3, k+6, k+7}
//   -> lane needs A[row][k+4h..k+4h+3] (one b128) and B[k+4h..+3][n] (one b128)
// C/D layout: acc[t][r] = element (M = r + 8*half, N = t*16 + l).
// NOTE: no early wave exit — every wave joins staging + barriers; EXEC stays
// all-ones for WMMA; only the final stores are guarded.
// ---------------------------------------------------------------------------
template<int NT, int CK>
__global__ void __launch_bounds__(256)
gemm_relu_wmma(const float* __restrict__ A, int lda,
               const float* __restrict__ Bq,    // quad-packed [K/4][NT*16][4]
               const float* __restrict__ bias,  // [NT*16] zero-padded
               float* __restrict__ out, int ldo, int out_cols,
               int K, int N, int relu)
{
    constexpr int LDQ    = NT * 16 * 4;          // floats per quad-row (4 k's)
    constexpr int CHUNKF = (CK / 4) * LDQ;       // floats per K-chunk of B
    constexpr int NLDR   = CHUNKF / (256 * 4);   // v4f per thread per stage
    extern __shared__ float smem[];              // 2 * CHUNKF floats

    const int tid   = threadIdx.x;
    const int lane  = tid & 31;
    const int wave  = tid >> 5;
    const int mbase = (blockIdx.x * 8 + wave) * 16;
    const int half  = lane >> 4;                 // 0 or 1
    const int l     = lane & 15;

    int arow = mbase + l;
    if (arow >= N) arow = N - 1;                 // clamp for inactive tail waves
    const float* __restrict__ Ap = A + (size_t)arow * lda + 4 * half;

    v8f acc[NT];
    v8f z8 = {0.f, 0.f, 0.f, 0.f, 0.f, 0.f, 0.f, 0.f};
#pragma unroll
    for (int t = 0; t < NT; ++t) acc[t] = z8;

    const int nchunk = K / CK;

    // ---- stage chunk 0 into LDS buffer 0 ----
#pragma unroll
    for (int i = 0; i < NLDR; ++i) {
        v4f v = *(const v4f*)(Bq + tid * 4 + i * 1024);
        *(v4f*)&smem[tid * 4 + i * 1024] = v;
    }
    __syncthreads();

    for (int c = 0; c < nchunk; ++c) {
        const int cur = c & 1;
        const bool more = (c + 1 < nchunk);      // block-uniform

        // issue global loads for the NEXT chunk early (latency-hidden by WMMA)
        v4f sreg[NLDR];
        if (more) {
            const float* __restrict__ G = Bq + (size_t)(c + 1) * CHUNKF;
#pragma unroll
            for (int i = 0; i < NLDR; ++i)
                sreg[i] = *(const v4f*)(G + tid * 4 + i * 1024);
        }

        // compute on current LDS buffer
        const float* __restrict__ Bl = &smem[cur * CHUNKF + half * LDQ + l * 4];
        const int kb = c * CK;
#pragma unroll 4
        for (int s = 0; s < CK; s += 8) {
            v4f a = *(const v4f*)(Ap + kb + s);            // A[row][k+4h..+3]
            v2f a01 = __builtin_shufflevector(a, a, 0, 1);
            v2f a23 = __builtin_shufflevector(a, a, 2, 3);
            const float* __restrict__ Bs = Bl + (s >> 2) * LDQ;
#pragma unroll
            for (int t = 0; t < NT; ++t) {
                v4f bq = *(const v4f*)(Bs + t * 64);       // ds_load_b128
                v2f b01 = __builtin_shufflevector(bq, bq, 0, 1);
                v2f b23 = __builtin_shufflevector(bq, bq, 2, 3);
                acc[t] = __builtin_amdgcn_wmma_f32_16x16x4_f32(
                    false, a01, false, b01, (short)0, acc[t], false, false);
                acc[t] = __builtin_amdgcn_wmma_f32_16x16x4_f32(
                    false, a23, false, b23, (short)0, acc[t], false, false);
            }
        }

        // commit next chunk to the other LDS buffer
        if (more) {
            float* __restrict__ Dl = &smem[(cur ^ 1) * CHUNKF];
#pragma unroll
            for (int i = 0; i < NLDR; ++i)
                *(v4f*)&Dl[tid * 4 + i * 1024] = sreg[i];
        }
        __syncthreads();
    }

    if (mbase < N) {                             // wave-uniform store guard
#pragma unroll
        for (int t = 0; t < NT; ++t) {
            int n = t * 16 + l;
            float bv = bias[n];
#pragma unroll
            for (int r = 0; r < 8; ++r) {
                int row = mbase + r + 8 * half;
                float v = acc[t][r] + bv;
                if (relu) v = fmaxf(v, 0.f);
                if (row < N && n < out_cols)
                    out[(size_t)row * ldo + n] = v;
            }
        }
    }
}

// Quad-pack src[src_rows x src_cols] into dst[Kpad/4][Ncols][4], zero-filled:
// dst[(k/4)*Ncols*4 + n*4 + (k&3)] = src[k][n]
__global__ void pack_quad(const float* __restrict__ src, float* __restrict__ dst,
                          int src_rows, int src_cols, int Kpad, int Ncols)
{
    int i = blockIdx.x * blockDim.x + threadIdx.x;
    int total = Kpad * Ncols;
    if (i >= total) return;
    int kq  = i / (Ncols * 4);
    int rem = i - kq * (Ncols * 4);
    int n   = rem >> 2;
    int j   = rem & 3;
    int k   = kq * 4 + j;
    dst[i] = (k < src_rows && n < src_cols) ? src[k * src_cols + n] : 0.f;
}

// Zero-padded 1D pack for biases.
__global__ void pack_vec(const float* __restrict__ src, float* __restrict__ dst,
                         int src_n, int dst_n)
{
    int i = blockIdx.x * blockDim.x + threadIdx.x;
    if (i >= dst_n) return;
    dst[i] = (i < src_n) ? src[i] : 0.f;
}

__global__ void copy_f32(const float* __restrict__ src, float* __restrict__ dst, int n)
{
    int i = blockIdx.x * blockDim.x + threadIdx.x;
    if (i < n) dst[i] = src[i];
}

// One KENN layer over edges: zout (pre-initialized to zin's values) receives
// clause boosts via f32 atomics. z arrays (~3.2 MB, stride ZS=8) are
// L2-resident; gathers are 2x b128 per endpoint.
__global__ void kenn_edges(const float* __restrict__ zin, float* __restrict__ zout,
                           const float* __restrict__ rel,
                           const int* __restrict__ sx, const int* __restrict__ sy,
                           const float* __restrict__ cw, int M)
{
    int e = blockIdx.x * blockDim.x + threadIdx.x;
    if (e >= M) return;
    const int ix = sx[e];
    const int iy = sy[e];
    const float a = -rel[e];
    const v4f* __restrict__ zx = (const v4f*)(zin + (size_t)ix * ZS);
    const v4f* __restrict__ zy = (const v4f*)(zin + (size_t)iy * ZS);
    v4f x0 = zx[0], x1 = zx[1];
    v4f y0 = zy[0], y1 = zy[1];
    float zxv[NC] = {x0.x, x0.y, x0.z, x0.w, x1.x, x1.y};
    float zyv[NC] = {y0.x, y0.y, y0.z, y0.w, y1.x, y1.y};
#pragma unroll
    for (int c = 0; c < NC; ++c) {
        float b = -zxv[c];
        float d =  zyv[c];
        float m = fmaxf(a, fmaxf(b, d));
        float ea = __expf(a - m);
        float eb = __expf(b - m);
        float ed = __expf(d - m);
        float w  = cw[c] / (ea + eb + ed);
        atomicAdd(&zout[(size_t)ix * ZS + c], -w * eb);   // d_x
        atomicAdd(&zout[(size_t)iy * ZS + c],  w * ed);   // d_y
    }
}

__global__ void softmax6_kernel(const float* __restrict__ z, float* __restrict__ out, int N)
{
    int i = blockIdx.x * blockDim.x + threadIdx.x;
    if (i >= N) return;
    const v4f* __restrict__ p = (const v4f*)(z + (size_t)i * ZS);
    v4f q0 = p[0], q1 = p[1];
    float v[NC] = {q0.x, q0.y, q0.z, q0.w, q1.x, q1.y};
    float m = -3.0e38f;
#pragma unroll
    for (int c = 0; c < NC; ++c) m = fmaxf(m, v[c]);
    float s = 0.f;
#pragma unroll
    for (int c = 0; c < NC; ++c) { v[c] = __expf(v[c] - m); s += v[c]; }
    float inv = 1.f / s;
#pragma unroll
    for (int c = 0; c < NC; ++c) out[(size_t)i * NC + c] = v[c] * inv;
}

extern "C" void kernel_launch(void* const* d_in, const int* in_sizes, int n_in,
                              void* d_out, int out_size, void* d_ws, size_t ws_size,
                              hipStream_t stream)
{
    const float* features  = (const float*)d_in[0];
    const float* relations = (const float*)d_in[1];
    const int*   sx        = (const int*)  d_in[2];
    const int*   sy        = (const int*)  d_in[3];
    const float* W1 = (const float*)d_in[4];
    const float* b1 = (const float*)d_in[5];
    const float* W2 = (const float*)d_in[6];
    const float* b2 = (const float*)d_in[7];
    const float* W3 = (const float*)d_in[8];
    const float* b3 = (const float*)d_in[9];
    const float* Wl = (const float*)d_in[10];
    const float* bl = (const float*)d_in[11];
    const float* cw1 = (const float*)d_in[12];
    const float* cw2 = (const float*)d_in[13];
    const float* cw3 = (const float*)d_in[14];

    const int D = 4096;                 // D_FEAT
    const int H = 50;                   // HIDDEN
    const int N = in_sizes[0] / D;      // nodes
    const int M = in_sizes[1];          // edges

    // ---- workspace layout (floats); all offsets multiples of 8 -> 32B align ----
    float* ws = (float*)d_ws;
    size_t off = 0;
    float* W1q = ws + off; off += (size_t)D * NPAD;      // quad-packed [D/4][64][4]
    float* W2q = ws + off; off += (size_t)NPAD * NPAD;   // [16][64][4]
    float* W3q = ws + off; off += (size_t)NPAD * NPAD;
    float* Wlq = ws + off; off += (size_t)NPAD * 16;     // [16][16][4]
    float* b1p = ws + off; off += NPAD;
    float* b2p = ws + off; off += NPAD;
    float* b3p = ws + off; off += NPAD;
    float* blp = ws + off; off += 16;
    float* xa  = ws + off; off += (size_t)N * NPAD;      // activations ping
    float* xb  = ws + off; off += (size_t)N * NPAD;      // activations pong
    float* za  = ws + off; off += (size_t)N * ZS;        // logits ping (stride 8)
    float* zb  = ws + off; off += (size_t)N * ZS;        // logits pong
    (void)ws_size; (void)n_in; (void)out_size;

    const int T = 256;
    auto blk = [](int total, int t) { return (total + t - 1) / t; };

    // ---- pack quad-interleaved weights / padded biases (tiny) ----
    pack_quad<<<blk(D * NPAD, T),    T, 0, stream>>>(W1, W1q, D, H, D,    NPAD);
    pack_quad<<<blk(NPAD * NPAD, T), T, 0, stream>>>(W2, W2q, H, H, NPAD, NPAD);
    pack_quad<<<blk(NPAD * NPAD, T), T, 0, stream>>>(W3, W3q, H, H, NPAD, NPAD);
    pack_quad<<<blk(NPAD * 16, T),   T, 0, stream>>>(Wl, Wlq, H, NC, NPAD, 16);
    pack_vec<<<1, T, 0, stream>>>(b1, b1p, H, NPAD);
    pack_vec<<<1, T, 0, stream>>>(b2, b2p, H, NPAD);
    pack_vec<<<1, T, 0, stream>>>(b3, b3p, H, NPAD);
    pack_vec<<<1, T, 0, stream>>>(bl, blp, NC, 16);

    // ---- MLP via fp32 WMMA (8 waves/block, one 16-row tile per wave) ----
    const int mtiles  = (N + 15) / 16;
    const int gblocks = (mtiles + 7) / 8;
    // dynamic LDS: 2 buffers of (CK/4)*NT*64 floats
    gemm_relu_wmma<4,128><<<gblocks, T, 2*128/4*4*64*4, stream>>>
        (features, D,    W1q, b1p, xa, NPAD, NPAD, D,    N, 1);   // 64 KB LDS
    gemm_relu_wmma<4,64><<<gblocks, T, 2*64/4*4*64*4, stream>>>
        (xa,       NPAD, W2q, b2p, xb, NPAD, NPAD, NPAD, N, 1);   // 32 KB LDS
    gemm_relu_wmma<4,64><<<gblocks, T, 2*64/4*4*64*4, stream>>>
        (xb,       NPAD, W3q, b3p, xa, NPAD, NPAD, NPAD, N, 1);
    gemm_relu_wmma<1,64><<<gblocks, T, 2*64/4*1*64*4, stream>>>
        (xa,       NPAD, Wlq, blp, za, ZS,   NC,   NPAD, N, 0);   // 8 KB LDS

    // ---- 3 KENN layers (relations reused each layer, as in reference) ----
    const int zelems = N * ZS;
    copy_f32  <<<blk(zelems, T), T, 0, stream>>>(za, zb, zelems);
    kenn_edges<<<blk(M, T),      T, 0, stream>>>(za, zb, relations, sx, sy, cw1, M);
    copy_f32  <<<blk(zelems, T), T, 0, stream>>>(zb, za, zelems);
    kenn_edges<<<blk(M, T),      T, 0, stream>>>(zb, za, relations, sx, sy, cw2, M);
    copy_f32  <<<blk(zelems, T), T, 0, stream>>>(za, zb, zelems);
    kenn_edges<<<blk(M, T),      T, 0, stream>>>(za, zb, relations, sx, sy, cw3, M);

    // ---- final softmax ----
    softmax6_kernel<<<blk(N, T), T, 0, stream>>>(zb, (float*)d_out, N);
}